// ChebyshevKANLayer_3728031613251
// MI455X (gfx1250) — compile-verified
//
#include <hip/hip_runtime.h>
#include <math.h>

typedef float v2f __attribute__((ext_vector_type(2)));
typedef float v8f __attribute__((ext_vector_type(8)));

#define KAN_BATCH 2048
#define KAN_IN    256
#define KAN_OUT   256
#define KAN_DEG   8
#define NELEM_X   (KAN_BATCH * KAN_IN)   // 524288

// fast swish: x * sigmoid(x) using v_rcp_f32 instead of IEEE divide
__device__ __forceinline__ float kan_swish(float v) {
    return v * __builtin_amdgcn_rcpf(1.0f + __expf(-v));
}

// ---------------------------------------------------------------------------
// Kernel 1: per-block min/max partial reduction of x
// ---------------------------------------------------------------------------
__global__ void kan_minmax_partial(const float* __restrict__ x,
                                   float* __restrict__ partial, int n) {
    __shared__ float smin[256];
    __shared__ float smax[256];
    int tid = threadIdx.x;
    float mn = 3.4e38f, mx = -3.4e38f;
    for (int i = blockIdx.x * blockDim.x + tid; i < n; i += gridDim.x * blockDim.x) {
        float v = x[i];
        mn = fminf(mn, v);
        mx = fmaxf(mx, v);
    }
    smin[tid] = mn; smax[tid] = mx;
    __syncthreads();
    for (int s = 128; s > 0; s >>= 1) {
        if (tid < s) {
            smin[tid] = fminf(smin[tid], smin[tid + s]);
            smax[tid] = fmaxf(smax[tid], smax[tid + s]);
        }
        __syncthreads();
    }
    if (tid == 0) {
        partial[2 * blockIdx.x]     = smin[0];
        partial[2 * blockIdx.x + 1] = smax[0];
    }
}

// ---------------------------------------------------------------------------
// Kernel 2: final reduce (512 pairs) -> affine params: xs = a*x + bias
// ---------------------------------------------------------------------------
__global__ void kan_minmax_final(const float* __restrict__ partial,
                                 float* __restrict__ params, int npairs) {
    __shared__ float smin[256];
    __shared__ float smax[256];
    int tid = threadIdx.x;
    float mn = 3.4e38f, mx = -3.4e38f;
    for (int i = tid; i < npairs; i += 256) {
        mn = fminf(mn, partial[2 * i]);
        mx = fmaxf(mx, partial[2 * i + 1]);
    }
    smin[tid] = mn; smax[tid] = mx;
    __syncthreads();
    for (int s = 128; s > 0; s >>= 1) {
        if (tid < s) {
            smin[tid] = fminf(smin[tid], smin[tid + s]);
            smax[tid] = fmaxf(smax[tid], smax[tid + s]);
        }
        __syncthreads();
    }
    if (tid == 0) {
        float lo = smin[0], hi = smax[0];
        float inv = 2.0f / (hi - lo);
        params[0] = inv;               // scale
        params[1] = -lo * inv - 1.0f;  // bias
    }
}

// ---------------------------------------------------------------------------
// Kernel 3: base_out = swish(x) @ Wb   (M=2048, N=256, K=256) via WMMA f32
// One wave per 16x16 C tile, 64 k-steps of V_WMMA_F32_16X16X4_F32.
// ---------------------------------------------------------------------------
__global__ void kan_base_gemm(const float* __restrict__ x,
                              const float* __restrict__ wb,
                              float* __restrict__ base_out) {
    int wave = threadIdx.x >> 5;
    int lane = threadIdx.x & 31;
    int task = blockIdx.x * 8 + wave;      // 0..2047
    int n_tile = task & 15;                // 16 tiles of N
    int b_tile = task >> 4;                // 128 tiles of M
    int m     = lane & 15;
    int khalf = lane >> 4;                 // 0 or 1
    int k0    = khalf * 2;                 // K offset within 4-slice
    int row   = b_tile * 16 + m;
    int n     = n_tile * 16 + (lane & 15);

    const float* xrow = x + (size_t)row * KAN_IN;
    v8f c = {};
    for (int kk = 0; kk < KAN_IN; kk += 4) {
        v2f a;
        a.x = kan_swish(xrow[kk + k0]);
        a.y = kan_swish(xrow[kk + k0 + 1]);
        v2f b;
        b.x = wb[(size_t)(kk + k0)     * KAN_OUT + n];
        b.y = wb[(size_t)(kk + k0 + 1) * KAN_OUT + n];
        c = __builtin_amdgcn_wmma_f32_16x16x4_f32(false, a, false, b,
                                                  (short)0, c, false, false);
    }
#pragma unroll
    for (int r = 0; r < 8; ++r) {
        base_out[(size_t)(b_tile * 16 + r + khalf * 8) * KAN_OUT + n] = c[r];
    }
}

// ---------------------------------------------------------------------------
// Kernel 4: out[b,i,o] = base[b,o] + sum_d T_d(xs[b,i]) * Wc[i,o,d]
// One wave handles (16-batch tile, one i), loops 16 o-tiles.
// base tile staged in LDS (shared by the 8 waves of a block -> same b-tile),
// and fed straight into the WMMA C operand: D = A*B + C is the final result.
// ---------------------------------------------------------------------------
__global__ void kan_cheb_main(const float* __restrict__ x,
                              const float* __restrict__ wc,
                              const float* __restrict__ base,
                              const float* __restrict__ params,
                              float* __restrict__ out) {
    __shared__ float base_lds[16 * KAN_OUT];  // 16 KB

    int wave = threadIdx.x >> 5;
    int lane = threadIdx.x & 31;
    int task = blockIdx.x * 8 + wave;   // 0..32767
    int i      = task & 255;
    int b_tile = task >> 8;             // 0..127  (same for all waves in block)
    int b_base = b_tile * 16;

    // cooperative 16x256 f32 base tile load (1024 float4 / 256 threads)
    const float4* bsrc = (const float4*)(base + (size_t)b_base * KAN_OUT);
    float4* bdst = (float4*)base_lds;
    for (int t = threadIdx.x; t < 1024; t += 256) bdst[t] = bsrc[t];
    __syncthreads();

    float scale = params[0];
    float bias  = params[1];

    int m     = lane & 15;
    int khalf = lane >> 4;
    int k0    = khalf * 2;

    // Chebyshev basis for this lane's batch row at feature i
    float xs = fmaf(scale, x[(size_t)(b_base + m) * KAN_IN + i], bias);
    float T[KAN_DEG];
    T[0] = 1.0f;
    T[1] = xs;
#pragma unroll
    for (int d = 2; d < KAN_DEG; ++d) T[d] = 2.0f * xs * T[d - 1] - T[d - 2];

    v2f A1; A1.x = T[k0];     A1.y = T[k0 + 1];     // K = 0..3
    v2f A2; A2.x = T[4 + k0]; A2.y = T[5 + k0];     // K = 4..7

    int n_l = lane & 15;
    float* obase = out + (size_t)b_base * (KAN_IN * KAN_OUT) + (size_t)i * KAN_OUT;

    for (int ot = 0; ot < 16; ++ot) {
        int o = ot * 16 + n_l;
        const float* wrow = wc + ((size_t)i * KAN_OUT + o) * KAN_DEG;
        v2f B1; B1.x = wrow[k0];     B1.y = wrow[k0 + 1];
        v2f B2; B2.x = wrow[4 + k0]; B2.y = wrow[5 + k0];

        v8f c;
#pragma unroll
        for (int r = 0; r < 8; ++r) c[r] = base_lds[(r + khalf * 8) * KAN_OUT + o];

        c = __builtin_amdgcn_wmma_f32_16x16x4_f32(false, A1, false, B1,
                                                  (short)0, c, false, false);
        c = __builtin_amdgcn_wmma_f32_16x16x4_f32(false, A2, false, B2,
                                                  (short)0, c, false, false);

#pragma unroll
        for (int r = 0; r < 8; ++r) {
            obase[(size_t)(r + khalf * 8) * (KAN_IN * KAN_OUT) + o] = c[r];
        }
    }
}

// ---------------------------------------------------------------------------
extern "C" void kernel_launch(void* const* d_in, const int* in_sizes, int n_in,
                              void* d_out, int out_size, void* d_ws, size_t ws_size,
                              hipStream_t stream) {
    const float* x  = (const float*)d_in[0];   // (2048, 256)
    const float* wb = (const float*)d_in[1];   // (256, 256)
    const float* wc = (const float*)d_in[2];   // (256, 256, 8)
    float* out = (float*)d_out;                // (2048, 256, 256)

    float* ws      = (float*)d_ws;
    float* partial = ws;            // 512 pairs = 1024 floats (4 KB)
    float* params  = ws + 1024;     // 2 floats
    float* base    = ws + 1088;     // 2048*256 floats (2 MB), 16B-aligned

    kan_minmax_partial<<<512, 256, 0, stream>>>(x, partial, NELEM_X);
    kan_minmax_final<<<1, 256, 0, stream>>>(partial, params, 512);
    kan_base_gemm<<<256, 256, 0, stream>>>(x, wb, base);        // 2048 wave tiles
    kan_cheb_main<<<4096, 256, 0, stream>>>(x, wc, base, params, out); // 32768 wave tasks
}